// HypergraphLearner_24275155157348
// MI455X (gfx1250) — compile-verified
//
#include <hip/hip_runtime.h>
#include <hip/hip_bf16.h>
#include <cfloat>
#include <cstdint>

// ---------------------------------------------------------------------------
// HypergraphLearner attention for MI455X (gfx1250, wave32).
// - All matrix math on V_WMMA_F32_16X16X4_F32 (full fp32 precision).
// - Attention main loop double-buffers K/V/K_aux tiles in LDS using
//   GLOBAL_LOAD_ASYNC_TO_LDS_B128/B64 (ASYNCcnt) to overlap fetch with WMMA.
// - Adjacency mask stream (134 MB, the HBM roofline term) prefetched ahead
//   via global_prefetch_b8.
// ---------------------------------------------------------------------------

typedef __attribute__((ext_vector_type(2))) float v2f;
typedef __attribute__((ext_vector_type(8))) float v8f;

#define BDIM  8
#define NDIM  2048
#define DDIM  128
#define DADIM 32
#define LROW  132   // 128 + 4 pad (row = 528B, 16B aligned, conflict-free)
#define LROWA 36    // 32 + 4 pad  (row = 144B, 16B aligned)
#define MASKV (-FLT_MAX)

__device__ __forceinline__ v8f wmma_f32(v2f a, v2f b, v8f c) {
  // D = A(16x4) * B(4x16) + C(16x16), fp32, wave32
  return __builtin_amdgcn_wmma_f32_16x16x4_f32(
      /*neg_a=*/false, a, /*neg_b=*/false, b,
      /*c_mod=*/(short)0, c, /*reuse_a=*/false, /*reuse_b=*/false);
}

// ---- CDNA5 async global->LDS copies (ASYNCcnt), GV addressing mode --------
// Low 32 bits of a generic pointer to __shared__ are the LDS byte offset
// (aperture mapping).
__device__ __forceinline__ void async_copy_b128(float* lds_dst, const float* g_src) {
  asm volatile("global_load_async_to_lds_b128 %0, %1, off"
               :: "v"((uint32_t)(uintptr_t)lds_dst), "v"(g_src) : "memory");
}
__device__ __forceinline__ void async_copy_b64(float* lds_dst, const float* g_src) {
  asm volatile("global_load_async_to_lds_b64 %0, %1, off"
               :: "v"((uint32_t)(uintptr_t)lds_dst), "v"(g_src) : "memory");
}
#define WAIT_ASYNCCNT(n) asm volatile("s_wait_asynccnt " #n ::: "memory")

// ---------------------------------------------------------------------------
// Projection: q = x @ Wq^T + bq (and k, v). One wave per 16x16 output tile.
// A-frag (16x4): lane(r,h) holds A[r][k0+2h], A[r][k0+1+2h].
// B-frag (4x16): lane(r,h) holds B[k0+2h][r], B[k0+1+2h][r]  (B = W^T -> W[r][k]).
// C/D (16x16):   lane(r,h) VGPR v holds C[v+8h][r].
// ---------------------------------------------------------------------------
__global__ __launch_bounds__(128) void hg_proj_kernel(
    const float* __restrict__ x,
    const float* __restrict__ Wq, const float* __restrict__ bq,
    const float* __restrict__ Wk, const float* __restrict__ bk,
    const float* __restrict__ Wv, const float* __restrict__ bv,
    float* __restrict__ ws) {
  const int lane = threadIdx.x & 31;
  const int wave = threadIdx.x >> 5;
  const int r = lane & 15;
  const int h = lane >> 4;

  const int nRowTiles = (BDIM * NDIM) / 16;  // 1024
  const int nColTiles = DDIM / 16;           // 8
  const int gw = blockIdx.x * 4 + wave;
  const int p  = gw / (nRowTiles * nColTiles);
  const int rm = gw % (nRowTiles * nColTiles);
  const int mt = rm / nColTiles;
  const int ct = rm % nColTiles;

  const float* W    = (p == 0) ? Wq : (p == 1) ? Wk : Wv;
  const float* bias = (p == 0) ? bq : (p == 1) ? bk : bv;
  float* dst = ws + (size_t)p * ((size_t)BDIM * NDIM * DDIM);

  const float* arow = x + (size_t)(mt * 16 + r) * DDIM;
  const float* brow = W + (size_t)(ct * 16 + r) * DDIM;

  v8f acc = {};
#pragma unroll
  for (int k0 = 0; k0 < DDIM; k0 += 4) {
    v2f a, b;
    a.x = arow[k0 + 2 * h];
    a.y = arow[k0 + 2 * h + 1];
    b.x = brow[k0 + 2 * h];
    b.y = brow[k0 + 2 * h + 1];
    acc = wmma_f32(a, b, acc);
  }

  const float bb = bias[ct * 16 + r];
#pragma unroll
  for (int v = 0; v < 8; ++v) {
    const int row = mt * 16 + v + 8 * h;
    dst[(size_t)row * DDIM + ct * 16 + r] = acc[v] + bb;
  }
}

// ---------------------------------------------------------------------------
// Flash attention with aux merge + adjacency mask.
// Block = 128 threads = 4 waves; block handles 64 rows, each wave 16 rows.
// K/V/K_aux column tiles double-buffered in LDS via async-to-LDS DMA.
// ---------------------------------------------------------------------------
__global__ __launch_bounds__(128) void hg_attn_kernel(
    const float* __restrict__ ws,          // q|k|v, each B*N*D
    const float* __restrict__ query_aux,   // B*N*DA
    const float* __restrict__ key_aux,     // B*N*DA
    const int*   __restrict__ adj,         // B*N*N
    const float* __restrict__ merge_coeff, // scalar
    const float* __restrict__ threshold,   // scalar
    float* __restrict__ out) {             // B*N*D
  __shared__ float qs[64 * LROW];
  __shared__ float qa[64 * LROWA];
  __shared__ float ks[2][16 * LROW];
  __shared__ float vs[2][16 * LROW];
  __shared__ float ka[2][16 * LROWA];
  __shared__ float pbuf[4][16][17];

  const int tid  = threadIdx.x;
  const int lane = tid & 31;
  const int wave = tid >> 5;
  const int r = lane & 15;
  const int h = lane >> 4;

  const int b      = blockIdx.x >> 5;          // N/64 = 32 row-blocks
  const int mBlock = (blockIdx.x & 31) * 64;
  const int mBase  = mBlock + wave * 16;       // this wave's 16 rows

  const size_t PLANE = (size_t)BDIM * NDIM * DDIM;
  const float* qg = ws;
  const float* kg = ws + PLANE;
  const float* vg = ws + 2 * PLANE;

  const float* kbatch  = kg + (size_t)b * NDIM * DDIM;
  const float* vbatch  = vg + (size_t)b * NDIM * DDIM;
  const float* kabatch = key_aux + (size_t)b * NDIM * DADIM;

  // per-thread slot indices for the async tile copies (uniform per wave:
  // exactly 10 async instructions per thread per tile -> ASYNCcnt += 10)
  const int kvJJ[4] = {  (tid + 0*128) >> 5, (tid + 1*128) >> 5,
                         (tid + 2*128) >> 5, (tid + 3*128) >> 5 };
  const int kvDD[4] = { ((tid + 0*128) & 31) * 4, ((tid + 1*128) & 31) * 4,
                        ((tid + 2*128) & 31) * 4, ((tid + 3*128) & 31) * 4 };
  const int kaJJ[2] = {  (tid + 0*128) >> 4, (tid + 1*128) >> 4 };
  const int kaDD[2] = { ((tid + 0*128) & 15) * 2, ((tid + 1*128) & 15) * 2 };

  // ---- stage q (64x128) and q_aux (64x32) once (synchronous) ----
  {
    const float* qb = qg + ((size_t)b * NDIM + mBlock) * DDIM;
    for (int i = tid; i < 64 * (DDIM / 4); i += 128) {
      const int lr = i >> 5, dd = (i & 31) * 4;
      *(float4*)&qs[lr * LROW + dd] = *(const float4*)(qb + (size_t)lr * DDIM + dd);
    }
    const float* qab = query_aux + ((size_t)b * NDIM + mBlock) * DADIM;
    for (int i = tid; i < 64 * (DADIM / 4); i += 128) {
      const int lr = i >> 3, dd = (i & 7) * 4;
      *(float4*)&qa[lr * LROWA + dd] = *(const float4*)(qab + (size_t)lr * DADIM + dd);
    }
  }

  // ---- prologue: issue async copy of tile 0 into buffer 0 ----
  {
    const float* kb  = kbatch;
    const float* vb  = vbatch;
    const float* kab = kabatch;
#pragma unroll
    for (int t = 0; t < 4; ++t) {
      async_copy_b128(&ks[0][kvJJ[t] * LROW + kvDD[t]], kb + (size_t)kvJJ[t] * DDIM + kvDD[t]);
      async_copy_b128(&vs[0][kvJJ[t] * LROW + kvDD[t]], vb + (size_t)kvJJ[t] * DDIM + kvDD[t]);
    }
#pragma unroll
    for (int t = 0; t < 2; ++t)
      async_copy_b64(&ka[0][kaJJ[t] * LROWA + kaDD[t]], kab + (size_t)kaJJ[t] * DADIM + kaDD[t]);
  }
  __syncthreads();

  const float inv_scale  = 0.08838834764831845f;  // 1/sqrt(128)
  const float inv_ascale = 0.17677669529663687f;  // 1/sqrt(32)
  const float mc  = merge_coeff[0];
  const float thr = threshold[0];

  float m_run[8], l_run[8];
  v8f o[8];
#pragma unroll
  for (int v = 0; v < 8; ++v) { m_run[v] = -FLT_MAX; l_run[v] = 0.0f; o[v] = (v8f){}; }

  const float* qrow  = &qs[(wave * 16 + r) * LROW];
  const float* qarow = &qa[(wave * 16 + r) * LROWA];
  const int NT = NDIM / 16;  // 128 column tiles

  for (int jt = 0; jt < NT; ++jt) {
    const int cur = jt & 1;

    // ---- issue async copy of next tile into the other buffer, then drain
    //      the current tile's 10 in-flight ops for this wave ----
    if (jt + 1 < NT) {
      const int nxt = cur ^ 1;
      const float* kb  = kbatch  + (size_t)(jt + 1) * 16 * DDIM;
      const float* vb  = vbatch  + (size_t)(jt + 1) * 16 * DDIM;
      const float* kab = kabatch + (size_t)(jt + 1) * 16 * DADIM;
#pragma unroll
      for (int t = 0; t < 4; ++t) {
        async_copy_b128(&ks[nxt][kvJJ[t] * LROW + kvDD[t]], kb + (size_t)kvJJ[t] * DDIM + kvDD[t]);
        async_copy_b128(&vs[nxt][kvJJ[t] * LROW + kvDD[t]], vb + (size_t)kvJJ[t] * DDIM + kvDD[t]);
      }
#pragma unroll
      for (int t = 0; t < 2; ++t)
        async_copy_b64(&ka[nxt][kaJJ[t] * LROWA + kaDD[t]], kab + (size_t)kaJJ[t] * DADIM + kaDD[t]);
      WAIT_ASYNCCNT(10);
    } else {
      WAIT_ASYNCCNT(0);
    }
    __syncthreads();  // all waves' current-tile data visible in LDS

    // ---- S = Q K^T  (B-frag element v: K[r][k0+v+2h], i.e. k^T) ----
    v8f s = {};
    const float* krow = &ks[cur][r * LROW];
#pragma unroll
    for (int k0 = 0; k0 < DDIM; k0 += 4) {
      v2f a, bb;
      a.x  = qrow[k0 + 2 * h];  a.y  = qrow[k0 + 2 * h + 1];
      bb.x = krow[k0 + 2 * h];  bb.y = krow[k0 + 2 * h + 1];
      s = wmma_f32(a, bb, s);
    }
    // ---- S_aux = Qa Ka^T ----
    v8f sa = {};
    const float* karow = &ka[cur][r * LROWA];
#pragma unroll
    for (int k0 = 0; k0 < DADIM; k0 += 4) {
      v2f a, bb;
      a.x  = qarow[k0 + 2 * h];  a.y  = qarow[k0 + 2 * h + 1];
      bb.x = karow[k0 + 2 * h];  bb.y = karow[k0 + 2 * h + 1];
      sa = wmma_f32(a, bb, sa);
    }

    // ---- merge + adjacency mask (+ prefetch 4 tiles = 256B ahead) ----
    float sv[8];
#pragma unroll
    for (int v = 0; v < 8; ++v) {
      float sc = s[v] * inv_scale;
      const float sx = sa[v] * inv_ascale;
      if ((sc > thr) && (sx != 0.0f)) sc = (1.0f - mc) * sc + mc * sx;
      const int* amp = adj + ((size_t)b * NDIM + (mBase + v + 8 * h)) * NDIM + jt * 16 + r;
      const int msk = *amp;
      __builtin_prefetch(amp + 64, 0, 0);  // speculative: OOB silently dropped
      sv[v] = (msk == 0) ? MASKV : sc;
    }

    // ---- online softmax (row reductions across 16 lanes per half-wave) ----
#pragma unroll
    for (int v = 0; v < 8; ++v) {
      float tmax = sv[v];
#pragma unroll
      for (int off = 1; off < 16; off <<= 1)
        tmax = fmaxf(tmax, __shfl_xor(tmax, off, 32));
      const float mnew  = fmaxf(m_run[v], tmax);
      const float alpha = __expf(m_run[v] - mnew);
      const float p     = __expf(sv[v] - mnew);
      float tsum = p;
#pragma unroll
      for (int off = 1; off < 16; off <<= 1)
        tsum += __shfl_xor(tsum, off, 32);
      l_run[v] = l_run[v] * alpha + tsum;
      m_run[v] = mnew;
#pragma unroll
      for (int t = 0; t < 8; ++t) o[t][v] = o[t][v] * alpha;
      pbuf[wave][v + 8 * h][r] = p;  // C/D layout -> row-major 16x16 in LDS
    }

    // ---- O += P @ V  (A-frag of P re-read row-major from per-wave LDS) ----
#pragma unroll
    for (int k0 = 0; k0 < 16; k0 += 4) {
      v2f a;
      a.x = pbuf[wave][r][k0 + 2 * h];
      a.y = pbuf[wave][r][k0 + 2 * h + 1];
#pragma unroll
      for (int t = 0; t < 8; ++t) {
        v2f bb;
        bb.x = vs[cur][(k0 + 2 * h)     * LROW + 16 * t + r];
        bb.y = vs[cur][(k0 + 2 * h + 1) * LROW + 16 * t + r];
        o[t] = wmma_f32(a, bb, o[t]);
      }
    }
    __syncthreads();  // everyone done reading buf `cur` before it is refilled
  }

  // ---- finalize: out = O / l ----
  float invl[8];
#pragma unroll
  for (int v = 0; v < 8; ++v) invl[v] = 1.0f / l_run[v];
#pragma unroll
  for (int t = 0; t < 8; ++t) {
#pragma unroll
    for (int v = 0; v < 8; ++v) {
      const int grow = mBase + v + 8 * h;
      out[((size_t)b * NDIM + grow) * DDIM + 16 * t + r] = o[t][v] * invl[v];
    }
  }
}

extern "C" void kernel_launch(void* const* d_in, const int* in_sizes, int n_in,
                              void* d_out, int out_size, void* d_ws, size_t ws_size,
                              hipStream_t stream) {
  const float* x    = (const float*)d_in[0];
  const float* qaux = (const float*)d_in[1];
  const float* kaux = (const float*)d_in[2];
  const int*   adj  = (const int*)d_in[3];
  const float* mc   = (const float*)d_in[4];
  const float* Wq   = (const float*)d_in[5];
  const float* bq   = (const float*)d_in[6];
  const float* Wk   = (const float*)d_in[7];
  const float* bk   = (const float*)d_in[8];
  const float* Wv   = (const float*)d_in[9];
  const float* bv   = (const float*)d_in[10];
  const float* thr  = (const float*)d_in[11];
  float* out = (float*)d_out;
  float* ws  = (float*)d_ws;  // q|k|v planes: 3 * B*N*D floats = 24 MB

  const int projWaves  = 3 * ((BDIM * NDIM) / 16) * (DDIM / 16);  // 24576
  const int projBlocks = projWaves / 4;                            // 6144
  hg_proj_kernel<<<projBlocks, 128, 0, stream>>>(x, Wq, bq, Wk, bk, Wv, bv, ws);

  const int attnBlocks = BDIM * (NDIM / 64);                       // 256
  hg_attn_kernel<<<attnBlocks, 128, 0, stream>>>(ws, qaux, kaux, adj, mc, thr, out);
}